// CRF_6262062317610
// MI455X (gfx1250) — compile-verified
//
#include <hip/hip_runtime.h>
#include <hip/hip_bf16.h>

#define CRF_B 2048
#define CRF_S 512
#define CRF_T 32

typedef float v2f __attribute__((ext_vector_type(2)));
typedef float v8f __attribute__((ext_vector_type(8)));

// D = A(16x4 f32) * B(4x16 f32) + C(16x16 f32), full-precision WMMA
__device__ __forceinline__ v8f wmma_f32x4(v2f a, v2f b, v8f c) {
    return __builtin_amdgcn_wmma_f32_16x16x4_f32(
        /*neg_a=*/false, a, /*neg_b=*/false, b,
        /*c_mod=*/(short)0, c, /*reuse_a=*/false, /*reuse_b=*/false);
}

// ---------------------------------------------------------------------------
// Kernel 1: gold path score per batch (sequential gather walk).
// Also emits per-batch sequence length (mask is a contiguous prefix).
// ---------------------------------------------------------------------------
__global__ __launch_bounds__(256)
void crf_score_kernel(const float* __restrict__ em,
                      const int* __restrict__ tags,
                      const unsigned char* __restrict__ mask,
                      const float* __restrict__ trans,
                      const float* __restrict__ start_trans,
                      const float* __restrict__ end_trans,
                      float* __restrict__ score_out,
                      int* __restrict__ len_out)
{
    int b = blockIdx.x * blockDim.x + threadIdx.x;
    if (b >= CRF_B) return;
    const float* emb = em + (size_t)b * CRF_S * CRF_T;
    const int* tgb = tags + (size_t)b * CRF_S;
    const unsigned char* mkb = mask + (size_t)b * CRF_S;

    int prev = tgb[0];
    float sc = start_trans[prev] + emb[prev];
    int len = 1;
    for (int s = 1; s < CRF_S; ++s) {
        int tg = tgb[s];
        int mk = mkb[s] ? 1 : 0;
        float mf = (float)mk;
        sc += (trans[prev * CRF_T + tg] + emb[(size_t)s * CRF_T + tg]) * mf;
        prev = tg;           // carry always (matches reference scan carry)
        len += mk;
    }
    sc += end_trans[tgb[len - 1]];
    score_out[b] = sc;
    len_out[b]   = len;
}

// ---------------------------------------------------------------------------
// Kernel 2: forward algorithm (partition function) via f32 WMMA with
// async-LDS double-buffered emission streaming.
// One wave32 per block, 16 batch rows per block.
//   alpha_new[b][j] = m_b + log( (exp(alpha[b]-m_b) @ exp(trans))[b][j] ) + em[b,s,j]
// exp(trans) B-fragments are built once (constant over all 512 steps).
// Emissions for step s+1 are DMA'd to LDS (GLOBAL_LOAD_ASYNC_TO_LDS_B128,
// ASYNCcnt) while step s's WMMA chain runs.
// ---------------------------------------------------------------------------
__global__ __launch_bounds__(32)
void crf_forward_kernel(const float* __restrict__ em,
                        const float* __restrict__ trans,
                        const float* __restrict__ start_trans,
                        const float* __restrict__ end_trans,
                        const float* __restrict__ score_in,
                        const int* __restrict__ len_in,
                        float* __restrict__ diff_out)
{
    __shared__ float sAlpha[16][33];   // alpha state, padded stride
    __shared__ float sP[16][33];       // exp(alpha - m)
    __shared__ float sM[16];           // per-batch running max
    __shared__ float sExpT[32][33];    // exp(transitions)
    __shared__ float sEm[2][16 * 32];  // double-buffered emission tiles (2KB each)

    const int lane  = threadIdx.x;          // 0..31
    const int bBase = blockIdx.x * 16;      // global batch base
    const int b     = lane & 15;            // batch row this lane serves
    const int half  = (lane < 16) ? 0 : 1;  // half-wave id
    const int nCol  = lane & 15;
    const int kb    = half ? 2 : 0;
    const int bOff  = half * 8;

    // exp(transitions) -> LDS (coalesced across lanes)
    #pragma unroll 4
    for (int i = 0; i < 32; ++i)
        sExpT[i][lane] = __expf(trans[i * CRF_T + lane]);

    // Per-lane constants for the async emission DMA.
    // Instruction q, lane l covers flat elements f = q*128 + l*4 .. +3 of the
    // [16 x 32] tile; element f maps to (bb = f>>5, j = f&31).
    unsigned voffq[4];   // per-lane global byte offset (batch + column part)
    unsigned ldsq[4];    // per-lane LDS byte address within buffer 0
    const unsigned ldsBase = (unsigned)(size_t)&sEm[0][0];
    #pragma unroll
    for (int q = 0; q < 4; ++q) {
        int f  = q * 128 + lane * 4;
        int bb = f >> 5;
        int jv = f & 31;
        voffq[q] = (unsigned)(bb * (CRF_S * CRF_T * 4) + jv * 4);
        ldsq[q]  = ldsBase + (unsigned)(f * 4);
    }

    // Per-lane sequence lengths for the 8 batch rows this lane writes back.
    int lenr[8];
    #pragma unroll
    for (int r = 0; r < 8; ++r)
        lenr[r] = len_in[bBase + r + bOff];

    // Kick off async DMA of the s=1 emission tile into buffer 1.
    {
        const unsigned long long sa =
            (unsigned long long)(em + ((size_t)bBase * CRF_S + 1) * CRF_T);
        #pragma unroll
        for (int q = 0; q < 4; ++q) {
            unsigned la = ldsq[q] + 2048u;  // buffer (1 & 1) == 1
            asm volatile("global_load_async_to_lds_b128 %0, %1, %2"
                         :: "v"(la), "v"(voffq[q]), "s"(sa) : "memory");
        }
    }
    __syncthreads();

    // Build constant B fragments (ISA 16x16x4 f32 layouts):
    //   B vgpr0: lanes0-15 -> (K=0, N=lane), lanes16-31 -> (K=2, N=lane-16)
    //   B vgpr1: lanes0-15 -> (K=1, N=lane), lanes16-31 -> (K=3, N=lane-16)
    v2f Bfrag[2][8];
    #pragma unroll
    for (int n = 0; n < 2; ++n)
        #pragma unroll
        for (int t = 0; t < 8; ++t) {
            Bfrag[n][t].x = sExpT[4 * t + kb + 0][16 * n + nCol];
            Bfrag[n][t].y = sExpT[4 * t + kb + 1][16 * n + nCol];
        }

    // alpha0[b][j] = start_trans[j] + em[b,0,j]  (each lane fills 16 entries)
    {
        const int jBase = half * 16;
        #pragma unroll 4
        for (int jj = 0; jj < 16; ++jj) {
            int j = jBase + jj;
            sAlpha[b][j] = start_trans[j] +
                           em[((size_t)(bBase + b) * CRF_S) * CRF_T + j];
        }
    }
    __syncthreads();

    for (int s = 1; s < CRF_S; ++s) {
        const int buf = s & 1;

        // Issue async DMA for step s+1 while we compute step s.
        // (Uniform branch: EXEC stays all-1s for the WMMAs below.)
        if (s + 1 < CRF_S) {
            const unsigned long long sa =
                (unsigned long long)(em + ((size_t)bBase * CRF_S + (size_t)(s + 1)) * CRF_T);
            const unsigned lb = ((unsigned)((s + 1) & 1)) * 2048u;
            #pragma unroll
            for (int q = 0; q < 4; ++q) {
                unsigned la = ldsq[q] + lb;
                asm volatile("global_load_async_to_lds_b128 %0, %1, %2"
                             :: "v"(la), "v"(voffq[q]), "s"(sa) : "memory");
            }
        }

        // ---- per-batch max of alpha (both half-waves compute same value) ----
        float m = sAlpha[b][0];
        #pragma unroll 8
        for (int j = 1; j < 32; ++j) m = fmaxf(m, sAlpha[b][j]);
        if (lane < 16) sM[b] = m;

        // ---- P = exp(alpha - m), each lane writes its 16 entries ----
        {
            const int jBase = half * 16;
            #pragma unroll 4
            for (int jj = 0; jj < 16; ++jj) {
                int j = jBase + jj;
                sP[b][j] = __expf(sAlpha[b][j] - m);
            }
        }
        __syncthreads();

        // ---- A fragments (ISA 16x4 f32 A layout):
        //   vgpr0: lanes0-15 (M=lane,K=0), lanes16-31 (M=lane-16,K=2)
        //   vgpr1: K=1 / K=3
        v2f A[8];
        #pragma unroll
        for (int t = 0; t < 8; ++t) {
            A[t].x = sP[b][4 * t + kb + 0];
            A[t].y = sP[b][4 * t + kb + 1];
        }

        // ---- 16 chained f32 WMMAs: acc[16x32] = P @ expT ----
        v8f acc0 = {};
        v8f acc1 = {};
        #pragma unroll
        for (int t = 0; t < 8; ++t) {
            acc0 = wmma_f32x4(A[t], Bfrag[0][t], acc0);
            acc1 = wmma_f32x4(A[t], Bfrag[1][t], acc1);
        }

        // Ensure step s's emission tile has landed in LDS.
        // After issuing s+1's group (4 instrs), ASYNCcnt<=4 => group s done
        // (async completions are in-order). Last step: drain to 0.
        if (s + 1 < CRF_S) {
            asm volatile("s_wait_asynccnt 4" ::: "memory");
        } else {
            asm volatile("s_wait_asynccnt 0" ::: "memory");
        }

        // ---- writeback: alpha_new = m_b + log(acc) + em, masked select ----
        // D layout: vgpr r, lanes0-15 -> (M=r, N=lane); lanes16-31 -> (M=r+8, N=lane-16)
        #pragma unroll
        for (int r = 0; r < 8; ++r) {
            const int bb = r + bOff;
            float mb = sM[bb];
            float e0 = sEm[buf][bb * 32 + nCol];
            float e1 = sEm[buf][bb * 32 + 16 + nCol];
            float a0 = mb + __logf(acc0[r]) + e0;
            float a1 = mb + __logf(acc1[r]) + e1;
            int   mk = (s < lenr[r]) ? 1 : 0;
            float old0 = sAlpha[bb][nCol];
            float old1 = sAlpha[bb][16 + nCol];
            sAlpha[bb][nCol]      = mk ? a0 : old0;
            sAlpha[bb][16 + nCol] = mk ? a1 : old1;
        }
        __syncthreads();
    }

    // ---- norm = logsumexp(alpha + end_trans); emit (norm - score) ----
    if (lane < 16) {
        float mx = sAlpha[b][0] + end_trans[0];
        #pragma unroll 8
        for (int j = 1; j < 32; ++j) mx = fmaxf(mx, sAlpha[b][j] + end_trans[j]);
        float sum = 0.0f;
        #pragma unroll 8
        for (int j = 0; j < 32; ++j) sum += __expf(sAlpha[b][j] + end_trans[j] - mx);
        float norm = mx + __logf(sum);
        diff_out[bBase + b] = norm - score_in[bBase + b];
    }
}

// ---------------------------------------------------------------------------
// Kernel 3: deterministic reduction of the 2048 per-batch terms -> scalar loss
// ---------------------------------------------------------------------------
__global__ __launch_bounds__(256)
void crf_reduce_kernel(const float* __restrict__ diff, float* __restrict__ out)
{
    __shared__ float sbuf[256];
    int t = threadIdx.x;
    float acc = 0.0f;
    for (int i = t; i < CRF_B; i += 256) acc += diff[i];
    sbuf[t] = acc;
    __syncthreads();
    for (int off = 128; off > 0; off >>= 1) {
        if (t < off) sbuf[t] += sbuf[t + off];
        __syncthreads();
    }
    if (t == 0) out[0] = sbuf[0] * (1.0f / (float)CRF_B);
}

extern "C" void kernel_launch(void* const* d_in, const int* in_sizes, int n_in,
                              void* d_out, int out_size, void* d_ws, size_t ws_size,
                              hipStream_t stream) {
    const float*         em    = (const float*)d_in[0];         // [B,S,T]
    const int*           tags  = (const int*)d_in[1];           // [B,S]
    const unsigned char* mask  = (const unsigned char*)d_in[2]; // [B,S] bool
    const float*         trans = (const float*)d_in[3];         // [T,T]
    const float*         stt   = (const float*)d_in[4];         // [T]
    const float*         ent   = (const float*)d_in[5];         // [T]

    float* score = (float*)d_ws;            // [B] floats
    float* diff  = score + CRF_B;           // [B] floats
    int*   lens  = (int*)(diff + CRF_B);    // [B] ints

    crf_score_kernel<<<CRF_B / 256, 256, 0, stream>>>(em, tags, mask, trans, stt, ent,
                                                      score, lens);
    crf_forward_kernel<<<CRF_B / 16, 32, 0, stream>>>(em, trans, stt, ent,
                                                      score, lens, diff);
    crf_reduce_kernel<<<1, 256, 0, stream>>>(diff, (float*)d_out);
}